// ShapeFunctionMapping_66365834658392
// MI455X (gfx1250) — compile-verified
//
#include <hip/hip_runtime.h>
#include <hip/hip_bf16.h>

#define GRID_N   128
#define NUM_CELLS (GRID_N * GRID_N * GRID_N)   // 2,097,152
#define INV_H    127.0f
#define CUTOFF   1e-9f
#define BLK      256

// ---------------------------------------------------------------------------
// Kernel 1: zero the float4 accumulation grid (graph-capture-safe)
// ---------------------------------------------------------------------------
__global__ void __launch_bounds__(BLK) zero_grid_kernel(float4* __restrict__ grid) {
    const int c = blockIdx.x * BLK + threadIdx.x;   // exact: NUM_CELLS/BLK blocks
    grid[c] = make_float4(0.0f, 0.0f, 0.0f, 0.0f);
}

// ---------------------------------------------------------------------------
// Kernel 2: P2G scatter. One lane per particle.
//   - particle_data staged LDS via CDNA5 async loads, overlapped with weight ALU
//   - 8 nodes x 4 channels of native global_atomic_add_f32 into L2-resident grid
// ---------------------------------------------------------------------------
__global__ void __launch_bounds__(BLK) p2g_kernel(const float* __restrict__ pos,
                                                  const float* __restrict__ dat,
                                                  const float* __restrict__ mass,
                                                  float*       __restrict__ grid,
                                                  int n) {
    __shared__ float sdat[BLK * 3];
    const int tid = threadIdx.x;
    const int i   = blockIdx.x * BLK + tid;
    const int j   = (i < n) ? i : (n - 1);          // clamp tail lanes (harmless loads)

    // --- async stage particle_data[j] (3 dwords) into LDS; inst_offset applies
    //     to BOTH the LDS and global address, so one base covers all 3 words ---
    const float*   gsrc = dat + (size_t)j * 3u;
    const unsigned lds  = (unsigned)(unsigned long long)(&sdat[tid * 3]); // low 32 bits = LDS offset
    asm volatile("global_load_async_to_lds_b32 %0, %1, off"          :: "v"(lds), "v"(gsrc) : "memory");
    asm volatile("global_load_async_to_lds_b32 %0, %1, off offset:4" :: "v"(lds), "v"(gsrc) : "memory");
    asm volatile("global_load_async_to_lds_b32 %0, %1, off offset:8" :: "v"(lds), "v"(gsrc) : "memory");

    // --- trilinear weights (overlaps with the async transfers above) ---
    const float bpx = pos[(size_t)j * 3 + 0] * INV_H;
    const float bpy = pos[(size_t)j * 3 + 1] * INV_H;
    const float bpz = pos[(size_t)j * 3 + 2] * INV_H;
    const float fx = floorf(bpx), fy = floorf(bpy), fz = floorf(bpz);
    const int   ix = (int)fx, iy = (int)fy, iz = (int)fz;
    const float tx = bpx - fx, ty = bpy - fy, tz = bpz - fz;
    float wx[2], wy[2], wz[2];
    wx[0] = 1.0f - tx; wx[1] = 1.0f - fabsf(tx - 1.0f);   // mirror ref: 1-|dist|
    wy[0] = 1.0f - ty; wy[1] = 1.0f - fabsf(ty - 1.0f);
    wz[0] = 1.0f - tz; wz[1] = 1.0f - fabsf(tz - 1.0f);
    const float m = mass[j];

    // consume staged data (each lane reads only its own slots -> wave-level wait ok)
    asm volatile("s_wait_asynccnt 0x0" ::: "memory");
    const float dx = sdat[tid * 3 + 0];
    const float dy = sdat[tid * 3 + 1];
    const float dz = sdat[tid * 3 + 2];

    if (i < n) {
        const int base = (ix * GRID_N + iy) * GRID_N + iz;
#pragma unroll
        for (int ox = 0; ox < 2; ++ox)
#pragma unroll
            for (int oy = 0; oy < 2; ++oy)
#pragma unroll
                for (int oz = 0; oz < 2; ++oz) {
                    const float w = wx[ox] * wy[oy] * wz[oz] * m;
                    float* cell = grid + 4u * (size_t)(base + (ox * GRID_N + oy) * GRID_N + oz);
                    // native f32 atomics (grid is coarse-grained hipMalloc memory)
                    unsafeAtomicAdd(cell + 0, w * dx);
                    unsafeAtomicAdd(cell + 1, w * dy);
                    unsafeAtomicAdd(cell + 2, w * dz);
                    unsafeAtomicAdd(cell + 3, w);
                }
    }
}

// ---------------------------------------------------------------------------
// Kernel 3: normalize grid in place (mask by weight cutoff, true division to
// match the reference's rounding)
// ---------------------------------------------------------------------------
__global__ void __launch_bounds__(BLK) normalize_kernel(float4* __restrict__ grid) {
    const int c = blockIdx.x * BLK + threadIdx.x;   // exact coverage
    float4 v = grid[c];
    if (v.w > CUTOFF) {
        v.x = v.x / v.w; v.y = v.y / v.w; v.z = v.z / v.w;
    } else {
        v.x = 0.0f; v.y = 0.0f; v.z = 0.0f;
    }
    grid[c] = v;
}

// ---------------------------------------------------------------------------
// Kernel 4: G2P gather. Recompute weights (ALU is free), 8x b128 gathers that
// hit the L2-resident grid, fma reduction, store [N,3].
// ---------------------------------------------------------------------------
__global__ void __launch_bounds__(BLK) g2p_kernel(const float*  __restrict__ pos,
                                                  const float4* __restrict__ grid,
                                                  float*        __restrict__ out,
                                                  int n) {
    const int i = blockIdx.x * BLK + threadIdx.x;
    if (i >= n) return;

    const float bpx = pos[(size_t)i * 3 + 0] * INV_H;
    const float bpy = pos[(size_t)i * 3 + 1] * INV_H;
    const float bpz = pos[(size_t)i * 3 + 2] * INV_H;
    const float fx = floorf(bpx), fy = floorf(bpy), fz = floorf(bpz);
    const int   ix = (int)fx, iy = (int)fy, iz = (int)fz;
    const float tx = bpx - fx, ty = bpy - fy, tz = bpz - fz;
    float wx[2], wy[2], wz[2];
    wx[0] = 1.0f - tx; wx[1] = 1.0f - fabsf(tx - 1.0f);
    wy[0] = 1.0f - ty; wy[1] = 1.0f - fabsf(ty - 1.0f);
    wz[0] = 1.0f - tz; wz[1] = 1.0f - fabsf(tz - 1.0f);

    const int base = (ix * GRID_N + iy) * GRID_N + iz;
    float ax = 0.0f, ay = 0.0f, az = 0.0f;
#pragma unroll
    for (int ox = 0; ox < 2; ++ox)
#pragma unroll
        for (int oy = 0; oy < 2; ++oy)
#pragma unroll
            for (int oz = 0; oz < 2; ++oz) {
                const float  w = wx[ox] * wy[oy] * wz[oz];
                const float4 g = grid[base + (ox * GRID_N + oy) * GRID_N + oz];
                ax = fmaf(w, g.x, ax);
                ay = fmaf(w, g.y, ay);
                az = fmaf(w, g.z, az);
            }
    out[(size_t)i * 3 + 0] = ax;
    out[(size_t)i * 3 + 1] = ay;
    out[(size_t)i * 3 + 2] = az;
}

// ---------------------------------------------------------------------------
extern "C" void kernel_launch(void* const* d_in, const int* in_sizes, int n_in,
                              void* d_out, int out_size, void* d_ws, size_t ws_size,
                              hipStream_t stream) {
    const float* pos  = (const float*)d_in[0];   // [N,3]
    const float* dat  = (const float*)d_in[1];   // [N,3]
    const float* mass = (const float*)d_in[2];   // [N]
    const int    n    = in_sizes[2];

    float4* grid = (float4*)d_ws;                // NUM_CELLS * 16B = 32 MB scratch

    const int pblocks = (n + BLK - 1) / BLK;
    const int gblocks = NUM_CELLS / BLK;         // 8192, exact

    zero_grid_kernel<<<gblocks, BLK, 0, stream>>>(grid);
    p2g_kernel<<<pblocks, BLK, 0, stream>>>(pos, dat, mass, (float*)grid, n);
    normalize_kernel<<<gblocks, BLK, 0, stream>>>(grid);
    g2p_kernel<<<pblocks, BLK, 0, stream>>>(pos, grid, (float*)d_out, n);
}